// FocalLossAdaptive_12627203851059
// MI455X (gfx1250) — compile-verified
//
#include <hip/hip_runtime.h>
#include <cstdint>
#include <math.h>
#include <float.h>

// ---------------------------------------------------------------------------
// Focal loss (gamma=3), size_average=False, over [N=8192, C=32000] f32 logits.
// Roofline: 8192*32000*4B = 1.05 GB read exactly once -> ~45us @ 23.3 TB/s;
// ~1.3 GFLOP + 0.33G v_exp_f32 is negligible -> purely HBM-bound.
// Strategy: one block per row, 320 threads (10 wave32s), single-pass online
// softmax; tiles streamed global->LDS with CDNA5 async copies (ASYNCcnt),
// 4-deep ring buffer (3 tiles in flight), one barrier per tile.
// ---------------------------------------------------------------------------

#define TPB    320
#define VEC    4
#define TILE   (TPB * VEC)     // 1280 floats per tile; 32000/1280 = 25 tiles
#define NBUF   4               // ring buffer depth (3 tiles in flight)
#define NWAVES (TPB / 32)      // 10 wave32s

typedef __attribute__((ext_vector_type(4))) int v4i;
typedef __attribute__((address_space(1))) v4i* gv4i_p;   // global int4*
typedef __attribute__((address_space(3))) v4i* lv4i_p;   // LDS int4*

__device__ __forceinline__ void async_copy_b128(void* lds_dst, const float* gsrc) {
#if __has_builtin(__builtin_amdgcn_global_load_async_to_lds_b128)
    // (global src, lds dst, imm offset, imm cpol)
    __builtin_amdgcn_global_load_async_to_lds_b128(
        (gv4i_p)(uintptr_t)gsrc,
        (lv4i_p)(uint32_t)(uintptr_t)lds_dst,
        0, 0);
#else
    uint32_t lds_off = (uint32_t)(uintptr_t)lds_dst;   // AS3 offset = low 32 bits
    asm volatile("global_load_async_to_lds_b128 %0, %1, off"
                 :: "v"(lds_off), "v"(gsrc)
                 : "memory");
#endif
}

__device__ __forceinline__ void wait_asynccnt_le(int rem) {
#if __has_builtin(__builtin_amdgcn_s_wait_asynccnt)
    if (rem >= 2)      __builtin_amdgcn_s_wait_asynccnt(2);
    else if (rem == 1) __builtin_amdgcn_s_wait_asynccnt(1);
    else               __builtin_amdgcn_s_wait_asynccnt(0);
#else
    if (rem >= 2)      asm volatile("s_wait_asynccnt 0x2" ::: "memory");
    else if (rem == 1) asm volatile("s_wait_asynccnt 0x1" ::: "memory");
    else               asm volatile("s_wait_asynccnt 0x0" ::: "memory");
#endif
}

// Combine two (max, sum-of-exp) softmax partials.
__device__ __forceinline__ void combine_ms(float& m, float& s, float mo, float so) {
    const float L2E = 1.4426950408889634f;
    float Mn = fmaxf(m, mo);
    s = s * exp2f((m - Mn) * L2E) + so * exp2f((mo - Mn) * L2E);
    m = Mn;
}

__global__ void __launch_bounds__(TPB)
focal_rows_kernel(const float* __restrict__ input,
                  const int*   __restrict__ target,
                  float* __restrict__ part,   // [N] per-row loss, or nullptr
                  float* __restrict__ out,    // scalar (atomic fallback path)
                  int C)
{
    __shared__ __align__(16) float tile[NBUF][TILE];
    __shared__ float red_m[NWAVES];
    __shared__ float red_s[NWAVES];

    const int   row  = blockIdx.x;
    const int   tid  = threadIdx.x;
    const float L2E  = 1.4426950408889634f;
    const float* rowp = input + (size_t)row * (size_t)C;
    const int   NT   = (C + TILE - 1) / TILE;

    // Issue one async b128 per lane per tile; addresses of ragged tail quads
    // are clamped in-range so the issue stays wave-uniform (ASYNCcnt lockstep)
    // and never faults. Clamped slots are never consumed.
    auto issue_tile = [&](int t) {
        int base = t * TILE + tid * VEC;
        int cb = base;
        if (cb + VEC > C) cb = (C >= VEC) ? (C - VEC) : 0;
        async_copy_b128(&tile[t & (NBUF - 1)][tid * VEC], rowp + cb);
    };

    // Prime the pipeline: up to 3 tiles in flight.
    for (int p = 0; p < NBUF - 1 && p < NT; ++p) issue_tile(p);

    float m = -FLT_MAX;
    float s = 0.0f;

    for (int t = 0; t < NT; ++t) {
        wait_asynccnt_le(NT - 1 - t < 2 ? NT - 1 - t : 2);  // tile t landed
        __syncthreads();   // all waves waited (tile t fully in LDS) AND all
                           // waves finished consuming tile t-1 (so buf
                           // (t+3)%4 == (t-1)%4 is free to overwrite).
        if (t + NBUF - 1 < NT) issue_tile(t + NBUF - 1);

        const int buf  = t & (NBUF - 1);
        const int base = t * TILE + tid * VEC;
        if (base + VEC <= C) {
            float4 v = *reinterpret_cast<const float4*>(&tile[buf][tid * VEC]);
            float mv = fmaxf(fmaxf(v.x, v.y), fmaxf(v.z, v.w));
            float Mn = fmaxf(m, mv);
            s = s * exp2f((m - Mn) * L2E)
              + exp2f((v.x - Mn) * L2E) + exp2f((v.y - Mn) * L2E)
              + exp2f((v.z - Mn) * L2E) + exp2f((v.w - Mn) * L2E);
            m = Mn;
        } else {
            // Ragged tail (never taken for C=32000): read straight from global.
            for (int k = 0; k < VEC; ++k) {
                int idx = base + k;
                if (idx < C) {
                    float x = rowp[idx];
                    float Mn = fmaxf(m, x);
                    s = s * exp2f((m - Mn) * L2E) + exp2f((x - Mn) * L2E);
                    m = Mn;
                }
            }
        }
    }

    // Intra-wave (wave32) reduction of (m, s).
    for (int off = 16; off; off >>= 1) {
        float mo = __shfl_xor(m, off, 32);
        float so = __shfl_xor(s, off, 32);
        combine_ms(m, s, mo, so);
    }
    const int lane = tid & 31;
    const int wid  = tid >> 5;
    if (lane == 0) { red_m[wid] = m; red_s[wid] = s; }
    __syncthreads();

    if (tid == 0) {
        float M = red_m[0], S = red_s[0];
        for (int w = 1; w < NWAVES; ++w) combine_ms(M, S, red_m[w], red_s[w]);

        const int tgt = target[row];
        float xt    = rowp[tgt];
        float logpt = xt - M - logf(S);           // log_softmax at target
        float pt    = exp2f(logpt * L2E);         // exp(logpt)
        float omp   = 1.0f - pt;
        float loss  = -(omp * omp * omp) * logpt; // gamma = 3
        if (part) part[row] = loss;
        else      atomicAdd(out, loss);
    }
}

// Deterministic single-block tree sum of the per-row losses.
__global__ void __launch_bounds__(256)
reduce_rows_kernel(const float* __restrict__ part, float* __restrict__ out, int n)
{
    __shared__ float sh[256];
    float acc = 0.0f;
    for (int i = threadIdx.x; i < n; i += 256) acc += part[i];
    sh[threadIdx.x] = acc;
    __syncthreads();
    for (int off = 128; off; off >>= 1) {
        if ((int)threadIdx.x < off) sh[threadIdx.x] += sh[threadIdx.x + off];
        __syncthreads();
    }
    if (threadIdx.x == 0) out[0] = sh[0];
}

extern "C" void kernel_launch(void* const* d_in, const int* in_sizes, int n_in,
                              void* d_out, int out_size, void* d_ws, size_t ws_size,
                              hipStream_t stream)
{
    const float* input  = (const float*)d_in[0];
    const int*   target = (const int*)d_in[1];
    float*       out    = (float*)d_out;

    const int N = in_sizes[1];                         // 8192 rows
    const int C = (int)((long long)in_sizes[0] / N);   // 32000 classes

    if (ws_size >= (size_t)N * sizeof(float)) {
        float* part = (float*)d_ws;
        focal_rows_kernel<<<dim3(N), dim3(TPB), 0, stream>>>(
            input, target, part, nullptr, C);
        reduce_rows_kernel<<<dim3(1), dim3(256), 0, stream>>>(part, out, N);
    } else {
        (void)hipMemsetAsync(out, 0, sizeof(float), stream);
        focal_rows_kernel<<<dim3(N), dim3(TPB), 0, stream>>>(
            input, target, nullptr, out, C);
    }
}